// GCNLayer_1219770712797
// MI455X (gfx1250) — compile-verified
//
#include <hip/hip_runtime.h>

#define NNODES 100000
#define NEDGES 1600000
#define FEATS 64
#define NTILES (NNODES / 16) // 6250, exact
#define BN_EPS 1e-5f

typedef float v2f __attribute__((ext_vector_type(2)));
typedef float v8f __attribute__((ext_vector_type(8)));

// ---------------------------------------------------------------- zero scratch
__global__ void zero_ws_kernel(float* __restrict__ ws, long n) {
    long g = (long)blockIdx.x * blockDim.x + threadIdx.x;
    long stride = (long)gridDim.x * blockDim.x;
    for (long i = g; i < n; i += stride) ws[i] = 0.0f;
}

// ------------------------------------------------------- edge scatter (SpMM)
// One thread per (edge, 4 feats). 102.4M hardware f32 atomics into L2-resident agg.
__global__ void edge_scatter_kernel(const float* __restrict__ feats,
                                    const int* __restrict__ src,
                                    const int* __restrict__ dst,
                                    float* __restrict__ agg) {
    long g = (long)blockIdx.x * blockDim.x + threadIdx.x;
    if (g >= (long)NEDGES * 16) return;
    int e = (int)(g >> 4);
    int q = ((int)g & 15) << 2;
    int s = src[e];
    int d = dst[e];
    const float4 v = *(const float4*)(feats + (long)s * FEATS + q);
    float* ap = agg + (long)d * FEATS + q;
    unsafeAtomicAdd(ap + 0, v.x);
    unsafeAtomicAdd(ap + 1, v.y);
    unsafeAtomicAdd(ap + 2, v.z);
    unsafeAtomicAdd(ap + 3, v.w);
}

// --------------------------------------------- fused dual-GEMM + relu + stats
// One wave32 per 16-row tile. h = relu(agg@W+b) + relu(feats@Wr+br), written to
// out; per-column sum/sumsq reduced in LDS then flushed with global atomics.
__global__ __launch_bounds__(256, 1)
void gcn_wmma_kernel(const float* __restrict__ agg,
                     const float* __restrict__ feats,
                     const float* __restrict__ Wmat,
                     const float* __restrict__ bvec,
                     const float* __restrict__ Wres,
                     const float* __restrict__ bres,
                     float* __restrict__ out,
                     float* __restrict__ colsum,
                     float* __restrict__ colsq) {
    __shared__ float s_sum[FEATS];
    __shared__ float s_sq[FEATS];
    int tid = threadIdx.x;
    if (tid < FEATS) { s_sum[tid] = 0.0f; s_sq[tid] = 0.0f; }
    __syncthreads();

    int lane = tid & 31;
    int wave = tid >> 5;
    int tile = blockIdx.x * 8 + wave;
    int lr   = lane & 15;   // column index within 16-wide tile / row for A frag
    int half = lane >> 4;   // 0: K=k0,k0+1 (rows 0-7 of C) ; 1: K=k0+2,k0+3 (rows 8-15)

    if (tile < NTILES) {    // wave-uniform guard: EXEC stays all-1s for WMMA
        int row0 = tile * 16;
        float hv[4][8];     // final h for 4 N-tiles x 8 C-registers

        #pragma unroll
        for (int br_i = 0; br_i < 2; ++br_i) {
            const float* A  = br_i ? feats : agg;
            const float* Wm = br_i ? Wres : Wmat;
            const float* bv = br_i ? bres : bvec;

            // A fragments: lane lr owns row row0+lr; K pair selected by half.
            v2f afrag[16];
            const float* arow = A + (long)(row0 + lr) * FEATS + 2 * half;
            #pragma unroll
            for (int k = 0; k < 16; ++k)
                afrag[k] = *(const v2f*)(arow + 4 * k);

            #pragma unroll
            for (int nt = 0; nt < 4; ++nt) {
                int col = nt * 16 + lr;
                v8f c = {0.f, 0.f, 0.f, 0.f, 0.f, 0.f, 0.f, 0.f};
                #pragma unroll
                for (int k = 0; k < 16; ++k) {
                    v2f bfrag;
                    bfrag.x = Wm[(4 * k + 2 * half) * FEATS + col];
                    bfrag.y = Wm[(4 * k + 2 * half + 1) * FEATS + col];
                    c = __builtin_amdgcn_wmma_f32_16x16x4_f32(
                            false, afrag[k], false, bfrag,
                            (short)0, c, false, false);
                }
                float bias = bv[col];
                #pragma unroll
                for (int r = 0; r < 8; ++r) {
                    float v = fmaxf(c[r] + bias, 0.0f);
                    if (br_i == 0) hv[nt][r] = v;
                    else           hv[nt][r] += v;
                }
            }
        }

        // store + per-column partial stats (all WMMA done; divergence OK now)
        #pragma unroll
        for (int nt = 0; nt < 4; ++nt) {
            int col = nt * 16 + lr;
            float s = 0.0f, sq = 0.0f;
            #pragma unroll
            for (int r = 0; r < 8; ++r) {
                int row = row0 + r + 8 * half;
                float v = hv[nt][r];
                out[(long)row * FEATS + col] = v;
                s  += v;
                sq += v * v;
            }
            // lanes l and l+16 hold the same column (different rows) -> combine
            s  += __shfl_xor(s, 16, 32);
            sq += __shfl_xor(sq, 16, 32);
            if (half == 0) {
                atomicAdd(&s_sum[col], s);   // ds_add_f32
                atomicAdd(&s_sq[col], sq);
            }
        }
    }

    __syncthreads();
    if (tid < FEATS) {
        unsafeAtomicAdd(&colsum[tid], s_sum[tid]);
        unsafeAtomicAdd(&colsq[tid], s_sq[tid]);
    }
}

// -------------------------------------------------------- batchnorm finalize
__global__ void bn_finalize_kernel(float* __restrict__ out,
                                   const float* __restrict__ colsum,
                                   const float* __restrict__ colsq,
                                   const float* __restrict__ gamma,
                                   const float* __restrict__ beta) {
    long g = (long)blockIdx.x * blockDim.x + threadIdx.x;
    if (g >= (long)NNODES * (FEATS / 4)) return;
    long i = g << 2;
    int c0 = (int)(i & (FEATS - 1));
    const float invN = 1.0f / (float)NNODES;
    float4 v = *(float4*)(out + i);
    float vv[4] = {v.x, v.y, v.z, v.w};
    float rr[4];
    #pragma unroll
    for (int j = 0; j < 4; ++j) {
        int c = c0 + j;
        float mean = colsum[c] * invN;
        float var  = colsq[c] * invN - mean * mean;
        float rstd = rsqrtf(var + BN_EPS);
        rr[j] = (vv[j] - mean) * rstd * gamma[c] + beta[c];
    }
    *(float4*)(out + i) = make_float4(rr[0], rr[1], rr[2], rr[3]);
}

// ----------------------------------------------------------------- launcher
extern "C" void kernel_launch(void* const* d_in, const int* in_sizes, int n_in,
                              void* d_out, int out_size, void* d_ws, size_t ws_size,
                              hipStream_t stream) {
    const float* feats = (const float*)d_in[0];
    const int*   src   = (const int*)d_in[1];
    const int*   dst   = (const int*)d_in[2];
    const float* W     = (const float*)d_in[3];
    const float* b     = (const float*)d_in[4];
    const float* Wr    = (const float*)d_in[5];
    const float* br    = (const float*)d_in[6];
    const float* gamma = (const float*)d_in[7];
    const float* beta  = (const float*)d_in[8];
    float* out = (float*)d_out;

    float* agg    = (float*)d_ws;                    // 100000*64 f32 = 25.6 MB
    float* colsum = agg + (size_t)NNODES * FEATS;    // 64 f32
    float* colsq  = colsum + FEATS;                  // 64 f32

    long zn = (long)NNODES * FEATS + 2 * FEATS;
    zero_ws_kernel<<<2048, 256, 0, stream>>>(agg, zn);

    long et = (long)NEDGES * 16;
    edge_scatter_kernel<<<(int)((et + 255) / 256), 256, 0, stream>>>(feats, src, dst, agg);

    gcn_wmma_kernel<<<(NTILES + 7) / 8, 256, 0, stream>>>(agg, feats, W, b, Wr, br,
                                                          out, colsum, colsq);

    long bt = (long)NNODES * (FEATS / 4);
    bn_finalize_kernel<<<(int)((bt + 255) / 256), 256, 0, stream>>>(out, colsum, colsq,
                                                                    gamma, beta);
}